// YoloLoss_82798379532476
// MI455X (gfx1250) — compile-verified
//
#include <hip/hip_runtime.h>
#include <stdint.h>

// ---- problem constants (from reference) ----
#define NC      20
#define B_      16
#define A_      5
#define H_      152
#define W_      152
#define T_      50
#define NCELLS  (B_*A_*H_*W_)          // 1,848,320 cells
#define CELLF   (6+NC)                 // 26 floats per cell
#define CELLB   (CELLF*4)              // 104 bytes per cell
#define IGNORE_T 0.5f
#define BADW     1.25f

// ---- dense streaming config (use the 320KB WGP LDS) ----
#define CHUNK_CELLS   1024
#define CHUNK_BYTES   (CHUNK_CELLS*CELLB)   // 106,496 B per buffer (x2 = 208 KB LDS)
#define NCHUNKS       (NCELLS/CHUNK_CELLS)  // 1805 (exact, no tail)
#define DENSE_GRID    361                   // 5 chunks per block
#define DENSE_THREADS 512                   // 16 waves; 2 cells/thread/chunk

typedef unsigned int u32x4 __attribute__((ext_vector_type(4)));
typedef int          i32x8 __attribute__((ext_vector_type(8)));
typedef int          i32x4 __attribute__((ext_vector_type(4)));

// ---------------- target decode (shared by prep/sparse) ----------------
__device__ __forceinline__ bool decode_tgt(const float* __restrict__ target,
                                           const int*   __restrict__ sizes,
                                           int b, int t,
                                           int& gi, int& gj, int& bn,
                                           float& gx, float& gy, float& gw, float& gh)
{
    const float* row = target + ((size_t)b*T_ + t)*(13+NC);
    gx = row[0]*0.25f; gy = row[1]*0.25f;
    gh = row[3]*0.25f; gw = row[4]*0.25f;          // note: gw from ch4, gh from ch3
    bool valid = (t < sizes[b]) && (gw > 0.f) && (gh > 0.f);
    int i = (int)gx; gi = i < 0 ? 0 : (i > W_-1 ? W_-1 : i);
    int j = (int)gy; gj = j < 0 ? 0 : (j > H_-1 ? H_-1 : j);
    const float awv[A_] = {4.f,8.f,16.f,24.f,12.f};
    const float ahv[A_] = {4.f,8.f,16.f,12.f,24.f};
    float agt = (gw+1.f)*(gh+1.f);
    float best = -1.f; bn = 0;
#pragma unroll
    for (int a = 0; a < A_; ++a) {
        float inter = (fminf(gw,awv[a])+1.f)*(fminf(gh,ahv[a])+1.f);
        float iou   = inter/(agt + (awv[a]+1.f)*(ahv[a]+1.f) - inter + 1e-16f);
        if (iou > best) { best = iou; bn = a; }
    }
    return valid;
}

// ---------------- kernel 0: zero workspace ----------------
__global__ void init_ws(unsigned int* __restrict__ w, int nwords)
{
    int i = blockIdx.x*blockDim.x + threadIdx.x;
    if (i < nwords) w[i] = 0u;
}

// ---------------- kernel 1: scatter mask / ignore flags ----------------
__global__ void prep_flags(const float* __restrict__ target,
                           const int*   __restrict__ sizes,
                           unsigned char* __restrict__ maskA,
                           unsigned char* __restrict__ ignA)
{
    int idx = blockIdx.x*blockDim.x + threadIdx.x;
    if (idx >= B_*T_) return;
    int b = idx / T_, t = idx % T_;
    int gi, gj, bn; float gx, gy, gw, gh;
    if (!decode_tgt(target, sizes, b, t, gi, gj, bn, gx, gy, gw, gh)) return;
    const float awv[A_] = {4.f,8.f,16.f,24.f,12.f};
    const float ahv[A_] = {4.f,8.f,16.f,12.f,24.f};
    float agt = (gw+1.f)*(gh+1.f);
#pragma unroll
    for (int a = 0; a < A_; ++a) {
        float inter = (fminf(gw,awv[a])+1.f)*(fminf(gh,ahv[a])+1.f);
        float iou   = inter/(agt + (awv[a]+1.f)*(ahv[a]+1.f) - inter + 1e-16f);
        if (iou > IGNORE_T)
            ignA[(((size_t)b*A_+a)*H_+gj)*W_+gi] = 1;
    }
    maskA[(((size_t)b*A_+bn)*H_+gj)*W_+gi] = 1;
}

// ---------------- kernel 2: sparse per-target losses (xywh MSE + class CE) ----
__global__ void sparse_loss(const float* __restrict__ pred,
                            const float* __restrict__ target,
                            const int*   __restrict__ sizes,
                            float* __restrict__ acc)
{
    int idx = blockIdx.x*blockDim.x + threadIdx.x;
    if (idx >= B_*T_) return;
    int b = idx / T_, t = idx % T_;
    int gi, gj, bn; float gx, gy, gw, gh;
    if (!decode_tgt(target, sizes, b, t, gi, gj, bn, gx, gy, gw, gh)) return;
    // dedup: highest-t target owning the cell wins (scatter last-write-wins)
    for (int t2 = t+1; t2 < T_; ++t2) {
        int gi2, gj2, bn2; float a0,a1,a2,a3;
        if (decode_tgt(target, sizes, b, t2, gi2, gj2, bn2, a0,a1,a2,a3) &&
            gi2 == gi && gj2 == gj && bn2 == bn) return;
    }
    const float* p = pred + (size_t)((((size_t)b*A_+bn)*H_+gj)*W_+gi)*CELLF;
    float x = p[1], y = p[2], h = p[4], w = p[5];
    const float awv[A_] = {4.f,8.f,16.f,24.f,12.f};
    const float ahv[A_] = {4.f,8.f,16.f,12.f,24.f};
    float txv = atanhf(gx - ((float)gi + 0.5f));
    float tyv = atanhf(gy - ((float)gj + 0.5f));
    float twv = logf(gw/awv[bn] + 1e-16f);
    float thv = logf(gh/ahv[bn] + 1e-16f);
    float s4 = (x-txv)*(x-txv) + (y-tyv)*(y-tyv)
             + (w-twv)*(w-twv) + (h-thv)*(h-thv);
    // class cross entropy at this cell
    const float* oh = target + ((size_t)b*T_ + t)*(13+NC) + 13;
    int label = 0; float bl = oh[0];
#pragma unroll
    for (int k = 1; k < NC; ++k) if (oh[k] > bl) { bl = oh[k]; label = k; }
    float m = p[6];
#pragma unroll
    for (int k = 1; k < NC; ++k) m = fmaxf(m, p[6+k]);
    float se = 0.f;
#pragma unroll
    for (int k = 0; k < NC; ++k) se += expf(p[6+k] - m);
    float ce = (m + logf(se)) - p[6+label];
    atomicAdd(&acc[4], s4);
    atomicAdd(&acc[5], ce);
}

// ---------------- TDM issue: DMA one contiguous chunk to LDS ----------------
__device__ __forceinline__ void tdm_issue(const void* gptr, unsigned lds_byte_off)
{
    unsigned long long ga = (unsigned long long)(uintptr_t)gptr;
    // D# group 0: count=1 | lds_addr | global_addr[56:0] | type=2 (bits 127:126)
    unsigned g00 = 1u;
    unsigned g01 = (unsigned)__builtin_amdgcn_readfirstlane((int)lds_byte_off);
    unsigned g02 = (unsigned)__builtin_amdgcn_readfirstlane((int)(unsigned)ga);
    unsigned g03 = (unsigned)__builtin_amdgcn_readfirstlane(
                       (int)(((unsigned)((ga>>32)&0x01FFFFFFull)) | 0x80000000u));
    u32x4 g0 = { g00, g01, g02, g03 };
    // D# group 1: data_size=3 (8B), tensor_dim0 large (no OOB), tensor_dim1=1,
    // tile_dim0=CHUNK_BYTES/8, tile_dim1=1, tile_dim2=0, stride0=tile_dim0
    const unsigned TD0   = 0x00FFFFFFu;
    const unsigned TILE0 = CHUNK_BYTES/8u;     // 13,312 x 8B = 106,496 B
    i32x8 g1 = { (int)(3u<<16),
                 (int)((TD0 & 0xFFFFu) << 16),
                 (int)((TD0 >> 16) | (1u << 16)),
                 (int)(TILE0 << 16),
                 1,
                 (int)TILE0,
                 0, 0 };
    i32x4 z4 = {0,0,0,0};
    i32x8 z8 = {0,0,0,0,0,0,0,0};
    __builtin_amdgcn_tensor_load_to_lds(g0, g1, z4, z4, z8, 0);
}

// ---------------- kernel 3: dense conf BCE stream (TDM double-buffered) ------
__global__ void __launch_bounds__(DENSE_THREADS)
dense_conf(const float* __restrict__ pred,
           const unsigned char* __restrict__ maskA,
           const unsigned char* __restrict__ ignA,
           float* __restrict__ acc)
{
    __shared__ __align__(16) unsigned char smem[2][CHUNK_BYTES];
    float s_noobj = 0.f, n_noobj = 0.f, s_obj = 0.f, n_obj = 0.f;
    const bool wave0 = (threadIdx.x < 32);

    int  c   = blockIdx.x;
    bool has = (c < NCHUNKS);
    if (wave0 && has)
        tdm_issue((const char*)pred + (size_t)c*CHUNK_BYTES,
                  (unsigned)(uintptr_t)&smem[0][0]);
    int cur = 0;
    while (has) {
        int  nc      = c + (int)gridDim.x;
        bool hasNext = (nc < NCHUNKS);
        if (wave0) {
            if (hasNext) {
                tdm_issue((const char*)pred + (size_t)nc*CHUNK_BYTES,
                          (unsigned)(uintptr_t)&smem[cur^1][0]);
                __builtin_amdgcn_s_wait_tensorcnt(1);
            } else {
                __builtin_amdgcn_s_wait_tensorcnt(0);
            }
        }
        __syncthreads();
#pragma unroll
        for (int k = 0; k < CHUNK_CELLS/DENSE_THREADS; ++k) {
            int local = threadIdx.x + k*DENSE_THREADS;
            int g     = c*CHUNK_CELLS + local;
            const float* lp = (const float*)&smem[cur][(size_t)local*CELLB];
            float conf = lp[0];                       // channel 0 (dword stride 26 -> 32 banks)
            float bce0 = fmaxf(conf, 0.f) + log1pf(expf(-fabsf(conf)));
            unsigned char m  = maskA[g];
            unsigned char ig = ignA[g];
            if (m)            { s_obj   += bce0 - conf; n_obj   += 1.f; }
            else if (!ig)     { s_noobj += bce0;        n_noobj += 1.f; }
        }
        __syncthreads();
        c = nc; has = hasNext; cur ^= 1;
    }
    // wave32 reduction, then one atomic per wave
#pragma unroll
    for (int o = 16; o > 0; o >>= 1) {
        s_noobj += __shfl_xor(s_noobj, o);
        n_noobj += __shfl_xor(n_noobj, o);
        s_obj   += __shfl_xor(s_obj,   o);
        n_obj   += __shfl_xor(n_obj,   o);
    }
    if ((threadIdx.x & 31) == 0) {
        atomicAdd(&acc[0], s_noobj);
        atomicAdd(&acc[1], n_noobj);
        atomicAdd(&acc[2], s_obj);
        atomicAdd(&acc[3], n_obj);
    }
}

// ---------------- kernel 4: finalize ----------------
__global__ void finalize(const float* __restrict__ acc, float* __restrict__ out)
{
    float n_obj   = fmaxf(acc[3], 1.f);
    float n_noobj = fmaxf(acc[1], 1.f);
    out[0] = (acc[4] + acc[2] + acc[5]) / n_obj + BADW * acc[0] / n_noobj;
}

// ---------------- host launch ----------------
extern "C" void kernel_launch(void* const* d_in, const int* in_sizes, int n_in,
                              void* d_out, int out_size, void* d_ws, size_t ws_size,
                              hipStream_t stream)
{
    const float* pred   = (const float*)d_in[0];
    const float* target = (const float*)d_in[1];
    const int*   sizes  = (const int*)d_in[2];
    float* out = (float*)d_out;

    float*         acc   = (float*)d_ws;                       // 16 floats
    unsigned char* maskA = (unsigned char*)d_ws + 64;          // NCELLS bytes
    unsigned char* ignA  = maskA + NCELLS;                     // NCELLS bytes

    const int nwords = (64 + 2*NCELLS) / 4;
    init_ws   <<<(nwords + 255)/256, 256, 0, stream>>>((unsigned int*)d_ws, nwords);
    prep_flags<<<(B_*T_ + 255)/256,  256, 0, stream>>>(target, sizes, maskA, ignA);
    sparse_loss<<<(B_*T_ + 255)/256, 256, 0, stream>>>(pred, target, sizes, acc);
    dense_conf<<<DENSE_GRID, DENSE_THREADS, 0, stream>>>(pred, maskA, ignA, acc);
    finalize  <<<1, 1, 0, stream>>>(acc, out);
}